// EvolveGCNH_Transformer_19696720019627
// MI455X (gfx1250) — compile-verified
//
#include <hip/hip_runtime.h>
#include <hip/hip_bf16.h>
#include <math.h>

// ---------------------------------------------------------------------------
// EvolveGCN-H + HyperConv + Transformer for MI455X (gfx1250), f32 end-to-end.
// All dense matmuls run on V_WMMA_F32_16X16X4_F32 (wave32). Each wave owns a
// 32x32 C macro-tile (2x2 WMMA tiles -> 4 wmma per 8 operand loads), with a
// branchless, address-clamped K loop (out-of-range rows/cols only pollute
// discarded C lanes; WMMA raises no exceptions) and software pipelining so
// loads for K+4 are in flight while the 4 WMMAs for K issue.
// Model is ~1.3 GFLOP / ~3 MB weights: fully L2-resident, latency-bound, so
// we keep reference f32 numerics instead of narrowing precision.
// ---------------------------------------------------------------------------

#define T_        3
#define NG_       90
#define Dm_       90
#define NTOT_     270
#define NSTATIC_  24300
#define NEDGE_    24660
#define NSTATLK_  24480   /* N_EDGE - TEMP_HALF */
#define TOPK_     180
#define DFF_      2048
#define NTOK_     721     /* 1 + 270 + 180 + 180 + 90 */

typedef float v2f __attribute__((ext_vector_type(2)));
typedef float v8f __attribute__((ext_vector_type(8)));

#define ACT_NONE 0
#define ACT_RELU 1
#define ACT_SIGM 2
#define ACT_TANH 3

__device__ __forceinline__ float act_apply(float v, int act) {
  if (act == ACT_RELU) return v > 0.f ? v : 0.f;
  if (act == ACT_SIGM) return 1.f / (1.f + expf(-v));
  if (act == ACT_TANH) return tanhf(v);
  return v;
}

__device__ __forceinline__ float mishf(float v) {
  float sp = (v > 20.f) ? v : log1pf(expf(v));
  return v * tanhf(sp);
}

__device__ __forceinline__ v8f wmma4(v2f a, v2f b, v8f c) {
  return __builtin_amdgcn_wmma_f32_16x16x4_f32(false, a, false, b,
                                               (short)0, c, false, false);
}

// ---------------------------------------------------------------------------
// WMMA f32 GEMM:  C = act( alpha * A(MxK) @ op(B) + bias + R1 + R2 )
// TRANSB=0: B is KxN row-major.  TRANSB=1: B stored NxK row-major (B^T used).
// One wave per 32x32 macro-tile, 4 waves per block.
// Fragment layouts per CDNA5 ISA 7.12.2:
//   A 16x4 f32: lane m=lane&15, holds K = k0+2*(lane>=16) + {0,1}
//   B 4x16    : lane n=lane&15, VGPR v holds K = v + 2*(lane>=16)  (same ks)
//   C/D 16x16 : VGPR r -> row r + 8*(lane>=16), col = lane&15
// ---------------------------------------------------------------------------
template <int TRANSB>
__global__ void wmma_gemm_f32(int M, int N, int K,
                              const float* __restrict__ A,
                              const float* __restrict__ B,
                              float* __restrict__ C,
                              const float* __restrict__ bias,
                              const float* __restrict__ R1,
                              const float* __restrict__ R2,
                              float alpha, int act) {
  const int lane   = threadIdx.x & 31;
  const int wave   = threadIdx.x >> 5;
  const int tilesN = (N + 31) >> 5;
  const int tilesM = (M + 31) >> 5;
  const int tile   = blockIdx.x * (blockDim.x >> 5) + wave;
  if (tile >= tilesM * tilesN) return;            // uniform per wave
  const int tm = (tile / tilesN) << 5;
  const int tn = (tile % tilesN) << 5;

  const int lid   = lane & 15;
  const int khalf = (lane >> 4) << 1;             // 0 or 2

  // clamped operand rows/cols (garbage only lands in discarded C lanes)
  const int ra0 = (tm + lid      < M) ? tm + lid      : M - 1;
  const int ra1 = (tm + 16 + lid < M) ? tm + 16 + lid : M - 1;
  const int ca0 = (tn + lid      < N) ? tn + lid      : N - 1;
  const int ca1 = (tn + 16 + lid < N) ? tn + 16 + lid : N - 1;

  const float* pa0 = A + (size_t)ra0 * K;
  const float* pa1 = A + (size_t)ra1 * K;
  const float* pb0 = TRANSB ? B + (size_t)ca0 * K : B + ca0;
  const float* pb1 = TRANSB ? B + (size_t)ca1 * K : B + ca1;

  auto loadf = [&](int kk, v2f& A0, v2f& A1, v2f& B0, v2f& B1) {
    const int ka = kk + khalf;
    A0.x = pa0[ka]; A0.y = pa0[ka + 1];
    A1.x = pa1[ka]; A1.y = pa1[ka + 1];
    if (TRANSB) {
      B0.x = pb0[ka]; B0.y = pb0[ka + 1];
      B1.x = pb1[ka]; B1.y = pb1[ka + 1];
    } else {
      B0.x = pb0[(size_t)ka * N]; B0.y = pb0[(size_t)(ka + 1) * N];
      B1.x = pb1[(size_t)ka * N]; B1.y = pb1[(size_t)(ka + 1) * N];
    }
  };

  v8f acc00 = {0.f,0.f,0.f,0.f,0.f,0.f,0.f,0.f};
  v8f acc01 = acc00, acc10 = acc00, acc11 = acc00;

  const int Kmain = K & ~3;                       // full 4-wide steps
  if (Kmain > 0) {
    v2f a0, a1, b0, b1;
    loadf(0, a0, a1, b0, b1);
    for (int k0 = 0; k0 < Kmain; k0 += 4) {
      const int kn = (k0 + 4 < Kmain) ? (k0 + 4) : 0;  // wrapped prefetch
      v2f na0, na1, nb0, nb1;
      loadf(kn, na0, na1, nb0, nb1);              // in flight during WMMAs
      acc00 = wmma4(a0, b0, acc00);
      acc01 = wmma4(a0, b1, acc01);
      acc10 = wmma4(a1, b0, acc10);
      acc11 = wmma4(a1, b1, acc11);
      a0 = na0; a1 = na1; b0 = nb0; b1 = nb1;
    }
  }
  if (Kmain < K) {                                // branchless masked K-tail
    const int ka  = Kmain + khalf;
    const float m0 = (ka     < K) ? 1.f : 0.f;
    const float m1 = (ka + 1 < K) ? 1.f : 0.f;
    const int kc0 = (ka     < K) ? ka     : K - 1;
    const int kc1 = (ka + 1 < K) ? ka + 1 : K - 1;
    v2f a0, a1, b0, b1;
    a0.x = pa0[kc0] * m0; a0.y = pa0[kc1] * m1;
    a1.x = pa1[kc0] * m0; a1.y = pa1[kc1] * m1;
    if (TRANSB) {
      b0.x = pb0[kc0] * m0; b0.y = pb0[kc1] * m1;
      b1.x = pb1[kc0] * m0; b1.y = pb1[kc1] * m1;
    } else {
      b0.x = pb0[(size_t)kc0 * N] * m0; b0.y = pb0[(size_t)kc1 * N] * m1;
      b1.x = pb1[(size_t)kc0 * N] * m0; b1.y = pb1[(size_t)kc1 * N] * m1;
    }
    acc00 = wmma4(a0, b0, acc00);
    acc01 = wmma4(a0, b1, acc01);
    acc10 = wmma4(a1, b0, acc10);
    acc11 = wmma4(a1, b1, acc11);
  }

  auto store_tile = [&](v8f acc, int rowbase, int col) {
#pragma unroll
    for (int r = 0; r < 8; ++r) {
      const int row = rowbase + r + ((lane >> 4) << 3);
      if (row < M && col < N) {
        float v = alpha * acc[r];
        if (bias) v += bias[col];
        if (R1)   v += R1[(size_t)row * N + col];
        if (R2)   v += R2[(size_t)row * N + col];
        C[(size_t)row * N + col] = act_apply(v, act);
      }
    }
  };
  store_tile(acc00, tm,      tn + lid);
  store_tile(acc01, tm,      tn + 16 + lid);
  store_tile(acc10, tm + 16, tn + lid);
  store_tile(acc11, tm + 16, tn + 16 + lid);
}

// ---------------------------------------------------------------------------
// Elementwise / small kernels
// ---------------------------------------------------------------------------
__global__ void k_zero(float* p, int n) {
  int i = blockIdx.x * blockDim.x + threadIdx.x;
  if (i < n) p[i] = 0.f;
}
__global__ void k_copy(float* o, const float* a, int n) {
  int i = blockIdx.x * blockDim.x + threadIdx.x;
  if (i < n) o[i] = a[i];
}
__global__ void k_mul(float* o, const float* a, const float* b, int n) {
  int i = blockIdx.x * blockDim.x + threadIdx.x;
  if (i < n) o[i] = a[i] * b[i];
}
__global__ void k_gru_update(float* Q, const float* u, const float* h, int n) {
  int i = blockIdx.x * blockDim.x + threadIdx.x;
  if (i < n) { float uu = u[i]; Q[i] = (1.f - uu) * Q[i] + uu * h[i]; }
}

// scores = x @ scorer / ||scorer|| + mask   (single block, 128 threads)
__global__ void k_grcu_score(const float* __restrict__ x,
                             const float* __restrict__ scorer,
                             const float* __restrict__ mask,
                             float* __restrict__ score) {
  __shared__ float red[128];
  int tid = threadIdx.x;
  float s = 0.f;
  for (int j = tid; j < NG_; j += 128) { float w = scorer[j]; s += w * w; }
  red[tid] = s; __syncthreads();
  for (int st = 64; st > 0; st >>= 1) {
    if (tid < st) red[tid] += red[tid + st];
    __syncthreads();
  }
  float nrm = sqrtf(red[0]);
  if (tid < NG_) {
    float dot = 0.f;
    for (int j = 0; j < NG_; ++j) dot += x[tid * NG_ + j] * scorer[j];
    score[tid] = dot / nrm + mask[tid];
  }
}

// z[:, rank(i)] = x[i,:] * tanh(score[i]); rank = stable descending sort
// position (jax.lax.top_k tie semantics: ties broken by lower index).
__global__ void k_grcu_build_z(const float* __restrict__ x,
                               const float* __restrict__ score,
                               float* __restrict__ z) {
  int i = threadIdx.x;
  if (i >= NG_) return;
  float si = score[i];
  int rank = 0;
  for (int j = 0; j < NG_; ++j) {
    float sj = score[j];
    if (sj > si || (sj == si && j < i)) rank++;
  }
  float t = tanhf(si);
  for (int j = 0; j < NG_; ++j) z[j * NG_ + rank] = x[i * NG_ + j] * t;
}

// ---------------------------------------------------------------------------
// Hypergraph conv. Edge attrs follow adjs_all row-major order; endpoints
// follow the concat([static, temporal]) order (reference quirk, replicated).
// edge_emb[e,d] = mish(u[e]*w[d] + b[d]) with u[e]=0.5*(s[src]+s[dst]),
// s[n] = (sum of incident edge attrs)/deg(n).
// ---------------------------------------------------------------------------
__device__ __forceinline__ float edge_attr_of(int e, const float* A) {
  if (e < 8190)        { int i = e / 91, p = e % 91;           // rows 0..89
    return (p < 90) ? A[i * 90 + p] : 1.f; }
  else if (e < 16470)  { int q = e - 8190; int i = q / 92, p = q % 92; // rows 90..179
    if (p == 0) return 1.f;
    if (p <= 90) return A[8100 + i * 90 + (p - 1)];
    return 1.f; }
  else                 { int q = e - 16470; int i = q / 91, p = q % 91; // rows 180..269
    if (p == 0) return 1.f;
    return A[16200 + i * 90 + (p - 1)]; }
}
__device__ __forceinline__ void edge_nodes_of(int e, int& s, int& d) {
  if (e < NSTATIC_) {
    int t = e / 8100, q = e % 8100;
    s = t * 90 + q / 90; d = t * 90 + q % 90;
  } else {
    int j = e - NSTATIC_;
    if (j < 90)       { s = j; d = j + 90; }
    else if (j < 270) { int i = 90 + (j - 90) / 2; s = i;
                        d = ((j - 90) & 1) ? i + 90 : i - 90; }
    else              { int i = 180 + (j - 270); s = i; d = i - 90; }
  }
}
__global__ void k_hyper_accum(const float* __restrict__ A,
                              float* ssum, float* bdeg) {
  int e = blockIdx.x * blockDim.x + threadIdx.x;
  if (e >= NEDGE_) return;
  float ea = edge_attr_of(e, A);
  int s, d; edge_nodes_of(e, s, d);
  atomicAdd(&ssum[s], ea); atomicAdd(&ssum[d], ea);
  atomicAdd(&bdeg[s], 1.f); atomicAdd(&bdeg[d], 1.f);
}
__global__ void k_compute_u(const float* ssum, const float* bdeg, float* u) {
  int e = blockIdx.x * blockDim.x + threadIdx.x;
  if (e >= NEDGE_) return;
  int s, d; edge_nodes_of(e, s, d);
  float sv = bdeg[s] > 0.f ? ssum[s] / bdeg[s] : 0.f;
  float dv = bdeg[d] > 0.f ? ssum[d] / bdeg[d] : 0.f;
  u[e] = 0.5f * (sv + dv);
}
// score[e] = tanh( mish(u*w+b) . pool_w / ||pool_w|| )  for static-like edges
__global__ void k_edge_score(const float* __restrict__ u,
                             const float* __restrict__ w,
                             const float* __restrict__ b,
                             const float* __restrict__ pw,
                             float* __restrict__ score) {
  int e = blockIdx.x * blockDim.x + threadIdx.x;
  if (e >= NSTATLK_) return;
  float nrm = 0.f;
  for (int d = 0; d < Dm_; ++d) nrm += pw[d] * pw[d];
  nrm = sqrtf(nrm);
  float ue = u[e], dot = 0.f;
  for (int d = 0; d < Dm_; ++d) dot += mishf(ue * w[d] + b[d]) * pw[d];
  score[e] = tanhf(dot / nrm);
}

// iterative argmax top-k (exact jax.lax.top_k order incl. tie-break)
__global__ void k_topk(float* score, int n, int k, float* vals, int* idxs) {
  __shared__ float sv[1024];
  __shared__ int   si[1024];
  for (int round = 0; round < k; ++round) {
    float best = -INFINITY; int bi = n;
    for (int j = threadIdx.x; j < n; j += blockDim.x) {
      float v = score[j];
      if (v > best || (v == best && j < bi)) { best = v; bi = j; }
    }
    sv[threadIdx.x] = best; si[threadIdx.x] = bi; __syncthreads();
    for (int s = blockDim.x >> 1; s > 0; s >>= 1) {
      if (threadIdx.x < s) {
        if (sv[threadIdx.x + s] > sv[threadIdx.x] ||
            (sv[threadIdx.x + s] == sv[threadIdx.x] &&
             si[threadIdx.x + s] < si[threadIdx.x])) {
          sv[threadIdx.x] = sv[threadIdx.x + s];
          si[threadIdx.x] = si[threadIdx.x + s];
        }
      }
      __syncthreads();
    }
    if (threadIdx.x == 0) {
      vals[round] = sv[0]; idxs[round] = si[0];
      score[si[0]] = -INFINITY;
    }
    __syncthreads();
  }
}

// tokens: [graph_emb | node_emb+tE0 | static_sel+tE1 | temporal+tE2 | ortho]
__global__ void k_build_tokens(float* __restrict__ tok,
                               const float* __restrict__ node_emb,
                               const float* __restrict__ u,
                               const float* __restrict__ w,
                               const float* __restrict__ b,
                               const int*   __restrict__ topi,
                               const float* __restrict__ topv,
                               const float* __restrict__ graph_emb,
                               const float* __restrict__ type_emb,
                               const float* __restrict__ ortho) {
  int t = blockIdx.x * blockDim.x + threadIdx.x;
  if (t >= NTOK_ * Dm_) return;
  int row = t / Dm_, d = t % Dm_;
  float v;
  if (row == 0)        v = graph_emb[d];
  else if (row <= 270) v = node_emb[(row - 1) * Dm_ + d] + type_emb[d];
  else if (row <= 450) {
    int k = row - 271; int e = topi[k];
    v = mishf(u[e] * w[d] + b[d]) * topv[k] + type_emb[Dm_ + d];
  } else if (row <= 630) {
    int e = NSTATLK_ + (row - 451);
    v = mishf(u[e] * w[d] + b[d]) + type_emb[2 * Dm_ + d];
  } else               v = ortho[(row - 631) * Dm_ + d];
  tok[t] = v;
}

// row softmax (block per row)
__global__ void k_softmax_rows(float* __restrict__ S, int C) {
  __shared__ float red[256];
  float* p = S + (size_t)blockIdx.x * C;
  float m = -INFINITY;
  for (int j = threadIdx.x; j < C; j += blockDim.x) m = fmaxf(m, p[j]);
  red[threadIdx.x] = m; __syncthreads();
  for (int s = blockDim.x >> 1; s > 0; s >>= 1) {
    if (threadIdx.x < s) red[threadIdx.x] = fmaxf(red[threadIdx.x], red[threadIdx.x + s]);
    __syncthreads();
  }
  m = red[0]; __syncthreads();
  float sum = 0.f;
  for (int j = threadIdx.x; j < C; j += blockDim.x) {
    float e = expf(p[j] - m); p[j] = e; sum += e;
  }
  red[threadIdx.x] = sum; __syncthreads();
  for (int s = blockDim.x >> 1; s > 0; s >>= 1) {
    if (threadIdx.x < s) red[threadIdx.x] += red[threadIdx.x + s];
    __syncthreads();
  }
  float inv = 1.f / red[0];
  for (int j = threadIdx.x; j < C; j += blockDim.x) p[j] *= inv;
}

// row LayerNorm (block per row), eps=1e-5
__global__ void k_layernorm(const float* __restrict__ X, float* __restrict__ Y,
                            const float* __restrict__ g, const float* __restrict__ b,
                            int C) {
  __shared__ float red[128];
  const float* x = X + (size_t)blockIdx.x * C;
  float* y = Y + (size_t)blockIdx.x * C;
  float s = 0.f;
  for (int j = threadIdx.x; j < C; j += blockDim.x) s += x[j];
  red[threadIdx.x] = s; __syncthreads();
  for (int st = 64; st > 0; st >>= 1) {
    if (threadIdx.x < st) red[threadIdx.x] += red[threadIdx.x + st];
    __syncthreads();
  }
  float mean = red[0] / C; __syncthreads();
  float v = 0.f;
  for (int j = threadIdx.x; j < C; j += blockDim.x) {
    float d = x[j] - mean; v += d * d;
  }
  red[threadIdx.x] = v; __syncthreads();
  for (int st = 64; st > 0; st >>= 1) {
    if (threadIdx.x < st) red[threadIdx.x] += red[threadIdx.x + st];
    __syncthreads();
  }
  float rstd = rsqrtf(red[0] / C + 1e-5f);
  for (int j = threadIdx.x; j < C; j += blockDim.x)
    y[j] = (x[j] - mean) * rstd * g[j] + b[j];
}

// out = mean_rows(h) @ cls_w.T + cls_b
__global__ void k_final(const float* __restrict__ h,
                        const float* __restrict__ clsw,
                        const float* __restrict__ clsb,
                        float* __restrict__ out) {
  __shared__ float red[128];
  float s = 0.f;
  if (threadIdx.x < Dm_) {
    for (int r = 0; r < NTOK_; ++r) s += h[r * Dm_ + threadIdx.x];
    s = (s / (float)NTOK_) * clsw[threadIdx.x];
  }
  red[threadIdx.x] = s; __syncthreads();
  for (int st = 64; st > 0; st >>= 1) {
    if (threadIdx.x < st) red[threadIdx.x] += red[threadIdx.x + st];
    __syncthreads();
  }
  if (threadIdx.x == 0) out[0] = red[0] + clsb[0];
}

// ---------------------------------------------------------------------------
// Host orchestration
// ---------------------------------------------------------------------------
// d_in layout (jax tree_leaves order = sorted dict keys at every level):
enum {
  IN_A = 0, IN_NODES = 1, IN_MASK = 2, IN_ORTHO = 3,
  P_CLS_B = 4, P_CLS_W = 5, P_GRAPH_EMB = 6,
  P_GRCU0 = 7,   // +0 Uh +1 Ur +2 Uu +3 Wh +4 Wr +5 Wu +6 bh +7 br +8 bu +9 gcn_init +10 scorer
  P_GRCU1 = 18,
  P_HCONV_B = 29, P_HCONV_W = 30,
  P_LAYER0 = 31, // +0 b1 +1 b2 +2 bk +3 bo +4 bq +5 bv +6 ln1_b +7 ln1_g +8 ln2_b +9 ln2_g +10 w1 +11 w2 +12 wk +13 wo +14 wq +15 wv
  P_LAYER1 = 47,
  P_LIN_B = 63, P_LIN_W = 64, P_POOL_W = 65, P_TYPE_EMB = 66
};

extern "C" void kernel_launch(void* const* d_in, const int* in_sizes, int n_in,
                              void* d_out, int out_size, void* d_ws, size_t ws_size,
                              hipStream_t stream) {
  (void)in_sizes; (void)n_in; (void)out_size; (void)ws_size;
  auto F = [&](int i) -> const float* { return (const float*)d_in[i]; };

  // ---- workspace carving (floats) ----
  float* W = (float*)d_ws;
  size_t off = 0;
  auto alloc = [&](size_t n) { float* p = W + off; off += n; return p; };
  float* nodesA = alloc(NSTATIC_);        // 3 x 90 x 90
  float* nodesB = alloc(NSTATIC_);
  float* Q    = alloc(8100);
  float* z    = alloc(8100);
  float* t1   = alloc(8100);
  float* upd  = alloc(8100);
  float* rst  = alloc(8100);
  float* RQ   = alloc(8100);
  float* hcap = alloc(8100);
  float* XQ   = alloc(8100);
  float* sc90 = alloc(128);
  float* u    = alloc(NEDGE_);
  float* ssum = alloc(NTOT_);
  float* bdeg = alloc(NTOT_);
  float* esc  = alloc(NSTATLK_);
  float* topv = alloc(TOPK_);
  int*   topi = (int*)alloc(TOPK_);
  float* tok  = alloc((size_t)NTOK_ * Dm_);
  float* qb   = alloc((size_t)NTOK_ * Dm_);
  float* kb   = alloc((size_t)NTOK_ * Dm_);
  float* vb   = alloc((size_t)NTOK_ * Dm_);
  float* Smat = alloc((size_t)NTOK_ * NTOK_);
  float* attn = alloc((size_t)NTOK_ * Dm_);
  float* hb   = alloc((size_t)NTOK_ * Dm_);
  float* ff1  = alloc((size_t)NTOK_ * DFF_);

  auto gemm = [&](int M, int N, int K, const float* A, const float* B, int tB,
                  float* C, const float* bias, const float* R1, const float* R2,
                  float alpha, int act) {
    int tiles = ((M + 31) / 32) * ((N + 31) / 32);
    int blocks = (tiles + 3) / 4;                  // 4 waves / block
    if (tB)
      wmma_gemm_f32<1><<<blocks, 128, 0, stream>>>(M, N, K, A, B, C,
                                                   bias, R1, R2, alpha, act);
    else
      wmma_gemm_f32<0><<<blocks, 128, 0, stream>>>(M, N, K, A, B, C,
                                                   bias, R1, R2, alpha, act);
  };
  auto ew = [&](int n) { return dim3((n + 255) / 256); };

  const float* A_in = F(IN_A);
  const float* mask = F(IN_MASK);

  // ================= GRCU x2 =================
  const float* cur = F(IN_NODES);
  float* dsts[2] = {nodesA, nodesB};
  for (int l = 0; l < 2; ++l) {
    int P = l ? P_GRCU1 : P_GRCU0;
    const float *Uh = F(P+0), *Ur = F(P+1), *Uu = F(P+2);
    const float *Wh = F(P+3), *Wr = F(P+4), *Wu = F(P+5);
    const float *bh = F(P+6), *br = F(P+7), *bu = F(P+8);
    const float *gcn = F(P+9), *scorer = F(P+10);
    float* dst = dsts[l];

    k_copy<<<ew(8100), 256, 0, stream>>>(Q, gcn, 8100);
    for (int t = 0; t < T_; ++t) {
      const float* x = cur + (size_t)t * 8100;
      k_grcu_score<<<1, 128, 0, stream>>>(x, scorer, mask + t * NG_, sc90);
      k_grcu_build_z<<<1, 128, 0, stream>>>(x, sc90, z);
      // upd = sigmoid(Wu@z + Uu@Q + bu)
      gemm(90, 90, 90, Wu, z, 0, t1, nullptr, bu, nullptr, 1.f, ACT_NONE);
      gemm(90, 90, 90, Uu, Q, 0, upd, nullptr, t1, nullptr, 1.f, ACT_SIGM);
      // rst = sigmoid(Wr@z + Ur@Q + br)
      gemm(90, 90, 90, Wr, z, 0, t1, nullptr, br, nullptr, 1.f, ACT_NONE);
      gemm(90, 90, 90, Ur, Q, 0, rst, nullptr, t1, nullptr, 1.f, ACT_SIGM);
      // hcap = tanh(Wh@z + Uh@(rst*Q) + bh)
      k_mul<<<ew(8100), 256, 0, stream>>>(RQ, rst, Q, 8100);
      gemm(90, 90, 90, Wh, z, 0, t1, nullptr, bh, nullptr, 1.f, ACT_NONE);
      gemm(90, 90, 90, Uh, RQ, 0, hcap, nullptr, t1, nullptr, 1.f, ACT_TANH);
      // Q = (1-upd)*Q + upd*hcap
      k_gru_update<<<ew(8100), 256, 0, stream>>>(Q, upd, hcap, 8100);
      // out_t = relu(A_t @ (x @ Q))
      gemm(90, 90, 90, x, Q, 0, XQ, nullptr, nullptr, nullptr, 1.f, ACT_NONE);
      gemm(90, 90, 90, A_in + (size_t)t * 8100, XQ, 0,
           dst + (size_t)t * 8100, nullptr, nullptr, nullptr, 1.f, ACT_RELU);
    }
    cur = dst;
  }
  const float* node_emb = cur;                    // (270, 90)

  // ================= hypergraph conv + edge top-k =================
  k_zero<<<ew(NTOT_), 256, 0, stream>>>(ssum, NTOT_);
  k_zero<<<ew(NTOT_), 256, 0, stream>>>(bdeg, NTOT_);
  k_hyper_accum<<<ew(NEDGE_), 256, 0, stream>>>(A_in, ssum, bdeg);
  k_compute_u<<<ew(NEDGE_), 256, 0, stream>>>(ssum, bdeg, u);
  k_edge_score<<<ew(NSTATLK_), 256, 0, stream>>>(
      u, F(P_HCONV_W), F(P_HCONV_B), F(P_POOL_W), esc);
  k_topk<<<1, 1024, 0, stream>>>(esc, NSTATLK_, TOPK_, topv, topi);

  // ================= tokens + input projection =================
  k_build_tokens<<<ew(NTOK_ * Dm_), 256, 0, stream>>>(
      tok, node_emb, u, F(P_HCONV_W), F(P_HCONV_B), topi, topv,
      F(P_GRAPH_EMB), F(P_TYPE_EMB), F(IN_ORTHO));
  // h = tokens @ lin_w.T + lin_b
  gemm(NTOK_, Dm_, Dm_, tok, F(P_LIN_W), 1, hb, F(P_LIN_B),
       nullptr, nullptr, 1.f, ACT_NONE);

  // ================= transformer x2 =================
  const float scale = 1.f / sqrtf((float)Dm_);
  for (int l = 0; l < 2; ++l) {
    int P = l ? P_LAYER1 : P_LAYER0;
    const float *b1 = F(P+0), *b2 = F(P+1), *bk = F(P+2), *bo = F(P+3);
    const float *bq = F(P+4), *bv = F(P+5);
    const float *ln1b = F(P+6), *ln1g = F(P+7), *ln2b = F(P+8), *ln2g = F(P+9);
    const float *w1 = F(P+10), *w2 = F(P+11);
    const float *wk = F(P+12), *wo = F(P+13), *wq = F(P+14), *wv = F(P+15);

    gemm(NTOK_, Dm_, Dm_, hb, wq, 1, qb, bq, nullptr, nullptr, 1.f, ACT_NONE);
    gemm(NTOK_, Dm_, Dm_, hb, wk, 1, kb, bk, nullptr, nullptr, 1.f, ACT_NONE);
    gemm(NTOK_, Dm_, Dm_, hb, wv, 1, vb, bv, nullptr, nullptr, 1.f, ACT_NONE);
    // S = softmax(q @ k.T * scale)
    gemm(NTOK_, NTOK_, Dm_, qb, kb, 1, Smat, nullptr, nullptr, nullptr,
         scale, ACT_NONE);
    k_softmax_rows<<<NTOK_, 256, 0, stream>>>(Smat, NTOK_);
    // attn = S @ v
    gemm(NTOK_, Dm_, NTOK_, Smat, vb, 0, attn, nullptr, nullptr, nullptr,
         1.f, ACT_NONE);
    // h = LN1(h + attn @ wo.T + bo)
    gemm(NTOK_, Dm_, Dm_, attn, wo, 1, tok, bo, hb, nullptr, 1.f, ACT_NONE);
    k_layernorm<<<NTOK_, 128, 0, stream>>>(tok, hb, ln1g, ln1b, Dm_);
    // h = LN2(h + relu(h @ w1.T + b1) @ w2.T + b2)
    gemm(NTOK_, DFF_, Dm_, hb, w1, 1, ff1, b1, nullptr, nullptr, 1.f, ACT_RELU);
    gemm(NTOK_, Dm_, DFF_, ff1, w2, 1, tok, b2, hb, nullptr, 1.f, ACT_NONE);
    k_layernorm<<<NTOK_, 128, 0, stream>>>(tok, hb, ln2g, ln2b, Dm_);
  }

  // ================= pool + classifier =================
  k_final<<<1, 128, 0, stream>>>(hb, F(P_CLS_W), F(P_CLS_B), (float*)d_out);
}